// Subpixel_35940286333160
// MI455X (gfx1250) — compile-verified
//
#include <hip/hip_runtime.h>

// ---------------------------------------------------------------------------
// Subpixel / pixel-shuffle (r=2) on (32,256,256,12) f32 -> (32,512,512,3).
// Pure permutation, HBM-bandwidth bound (~201 MB total -> ~8.6us floor at
// 23.3 TB/s). Uses the CDNA5 async global<->LDS data-mover path so that every
// global transaction is a coalesced 16B/lane b128 transfer in both directions.
//
// Decomposition: one input row (n,i) = 256 px * 48 B = 12 KB maps to output
// rows 2i and 2i+1 of image n, which are CONTIGUOUS 12 KB in the output
// (flat offset rowIdx*3072 floats on both sides). So each 256-thread block
// does an in-LDS permutation of one 12 KB chunk.
// ---------------------------------------------------------------------------

// Probe-derived builtin param type: pointer-to-int4 in global (AS1) address
// space for the global side; LDS side cast to int4 AS3* (exact match or
// implicitly convertible to void AS3*).
typedef int v4i __attribute__((__vector_size__(16)));
#define GP(p) ((__attribute__((address_space(1))) v4i*)(void*)(p))
#define LP(p) ((__attribute__((address_space(3))) v4i*)(p))

#if __has_builtin(__builtin_amdgcn_global_load_async_to_lds_b128)
#define ASYNC_LOAD_B128(g, l) \
  __builtin_amdgcn_global_load_async_to_lds_b128(GP(g), LP(l), 0, 0)
#else
#define ASYNC_LOAD_B128(g, l)                                              \
  asm volatile("global_load_async_to_lds_b128 %0, %1, off"                 \
               :: "v"((unsigned)(unsigned long long)(l)),                  \
                  "v"((unsigned long long)(g))                             \
               : "memory")
#endif

#if __has_builtin(__builtin_amdgcn_global_store_async_from_lds_b128)
#define ASYNC_STORE_B128(g, l) \
  __builtin_amdgcn_global_store_async_from_lds_b128(GP(g), LP(l), 0, 0)
#else
#define ASYNC_STORE_B128(g, l)                                             \
  asm volatile("global_store_async_from_lds_b128 %1, %0, off"              \
               :: "v"((unsigned)(unsigned long long)(l)),                  \
                  "v"((unsigned long long)(g))                             \
               : "memory")
#endif

#if __has_builtin(__builtin_amdgcn_s_wait_asynccnt)
#define WAIT_ASYNC0() __builtin_amdgcn_s_wait_asynccnt(0)
#else
#define WAIT_ASYNC0() asm volatile("s_wait_asynccnt 0" ::: "memory")
#endif

__global__ __launch_bounds__(256)
void subpixel_r2_kernel(const float* __restrict__ in, float* __restrict__ out) {
    // Block = one (n, i) input row: 256 px * 12 ch = 3072 floats = 12 KB.
    __shared__ float lin[3072];   // input row, input channel order
    __shared__ float lout[3072];  // [0..1535] = out row 2i, [1536..3071] = row 2i+1

    const int tid = threadIdx.x;                 // 0..255
    const unsigned long long blk = blockIdx.x;   // rowIdx = n*256 + i
    const float* gin  = in  + blk * 3072ull;
    float*       gout = out + blk * 3072ull;

    // ---- Stage 1: async DMA global -> LDS, 16B per lane, fully coalesced ----
#pragma unroll
    for (int k = 0; k < 3; ++k) {
        ASYNC_LOAD_B128(gin + k * 1024 + tid * 4, &lin[k * 1024 + tid * 4]);
    }
    WAIT_ASYNC0();
    __syncthreads();

    // ---- Stage 2: per-pixel shuffle into output layout ----------------------
    // out[n,2i+r2,2j+r1,m] = x[n,i,j, m*4 + r1*2 + r2]
    const int j = tid;
    float x[12];
#pragma unroll
    for (int c = 0; c < 12; ++c) x[c] = lin[12 * j + c];   // 48B aligned -> b128 x3

    // row 2i  (r2=0): pixel 2j -> ch {0,4,8}; pixel 2j+1 -> ch {2,6,10}
    lout[6 * j + 0] = x[0];
    lout[6 * j + 1] = x[4];
    lout[6 * j + 2] = x[8];
    lout[6 * j + 3] = x[2];
    lout[6 * j + 4] = x[6];
    lout[6 * j + 5] = x[10];
    // row 2i+1 (r2=1): pixel 2j -> ch {1,5,9}; pixel 2j+1 -> ch {3,7,11}
    lout[1536 + 6 * j + 0] = x[1];
    lout[1536 + 6 * j + 1] = x[5];
    lout[1536 + 6 * j + 2] = x[9];
    lout[1536 + 6 * j + 3] = x[3];
    lout[1536 + 6 * j + 4] = x[7];
    lout[1536 + 6 * j + 5] = x[11];

    __syncthreads();  // includes DScnt drain -> lout visible to async engine

    // ---- Stage 3: async DMA LDS -> global, 16B per lane, fully coalesced ----
#pragma unroll
    for (int k = 0; k < 3; ++k) {
        ASYNC_STORE_B128(gout + k * 1024 + tid * 4, &lout[k * 1024 + tid * 4]);
    }
    WAIT_ASYNC0();  // LDS must stay live until the DMA has read it
}

extern "C" void kernel_launch(void* const* d_in, const int* in_sizes, int n_in,
                              void* d_out, int out_size, void* d_ws, size_t ws_size,
                              hipStream_t stream) {
    const float* x   = (const float*)d_in[0];
    float*       out = (float*)d_out;
    // Number of 12KB row-blocks: (B*a) = total_elems / (b*c) = N / 3072.
    const int rows = in_sizes[0] / 3072;  // 32*256 = 8192 for the reference shapes
    subpixel_r2_kernel<<<rows, 256, 0, stream>>>(x, out);
}